// Bulid_ADJ_89189290869323
// MI455X (gfx1250) — compile-verified
//
#include <hip/hip_runtime.h>
#include <hip/hip_bf16.h>

#define Bb 4
#define Nn 512
#define Cc 52
#define Uu 128
#define KDIM 56            // 52 aa channels + by + by2, padded (bins 0 and 55 stay zero); 56 = 14*4 -> exact K for f32 WMMA
#define THRESH 0.02f
#define EPS 1e-05f

typedef float v2f __attribute__((ext_vector_type(2)));
typedef float v8f __attribute__((ext_vector_type(8)));

// -------------------------------------------------------------------------
// Kernel 1: per-(b,row) pairwise channel scan + LDS histogram + degree.
// One 256-thread block handles one adjacency row (each thread: 2 columns).
// -------------------------------------------------------------------------
__global__ __launch_bounds__(256) void pair_kernel(
    const float* __restrict__ loc, const float* __restrict__ inten,
    const float* __restrict__ pmass, const float* __restrict__ mass_aa,
    float* __restrict__ adj, float* __restrict__ cnt, float* __restrict__ dinv)
{
    const int i = blockIdx.x;   // row index 0..N-1
    const int b = blockIdx.y;   // batch
    const int t = threadIdx.x;

    __shared__ float s_loc[Nn];
    __shared__ float s_m[Nn];
    __shared__ float s_cnt[KDIM];
    __shared__ float s_deg;

    for (int j = t; j < Nn; j += 256) {
        s_loc[j] = loc[(size_t)b * Nn + j];
        s_m[j]   = (inten[(size_t)b * Nn + j] > EPS) ? 1.0f : 0.0f;
    }
    if (t < KDIM) s_cnt[t] = 0.0f;
    if (t == 0)   s_deg = 0.0f;
    __syncthreads();

    // Uniform constants -> SGPRs
    float ma[Cc];
    #pragma unroll
    for (int c = 0; c < Cc; ++c) ma[c] = mass_aa[c];
    const float pm  = pmass[b];
    const float pm2 = 0.5f * pm;

    const float li = s_loc[i];
    const float mi = s_m[i];

    float local_deg = 0.0f;

    #pragma unroll
    for (int jj = 0; jj < Nn / 256; ++jj) {
        const int j = t + jj * 256;
        const float mask = mi * s_m[j];
        const float d = fabsf(s_loc[j] - li) * mask;   // peak_matrix
        const float s = (s_loc[j] + li) * mask;        // peak_matrix_by

        // argmin over 54 channels (first-min semantics: strict <)
        float best = fabsf(d - ma[0]);
        int   bidx = 1;
        #pragma unroll
        for (int c = 1; c < Cc; ++c) {
            const float tt = fabsf(d - ma[c]);
            if (tt < best) { best = tt; bidx = c + 1; }
        }
        {   // channel 52 -> index 53 : |by - pm|
            const float tt = fabsf(s - pm);
            if (tt < best) { best = tt; bidx = 53; }
        }
        {   // channel 53 -> index 54 : |by - pm/2|
            const float tt = fabsf(s - pm2);
            if (tt < best) { best = tt; bidx = 54; }
        }

        const float a = (best <= THRESH) ? 1.0f : 0.0f;
        adj[(((size_t)b * Nn) + i) * Nn + j] = a;
        if (a != 0.0f) {
            atomicAdd(&s_cnt[bidx], 1.0f);   // ds_add_f32 histogram
            local_deg += 1.0f;
        }
    }

    atomicAdd(&s_deg, local_deg);
    __syncthreads();

    if (t < KDIM) cnt[(((size_t)b * Nn) + i) * KDIM + t] = s_cnt[t];
    if (t == 0)   dinv[(size_t)b * Nn + i] = rsqrtf(s_deg + 1.0f); // +1 from eye
}

// -------------------------------------------------------------------------
// Kernel 2: adj_norm[b,i,j] = dinv_i * (adj + I) * dinv_j, float4 vectorized
// -------------------------------------------------------------------------
__global__ __launch_bounds__(256) void norm_kernel(
    const float* __restrict__ adj, const float* __restrict__ dinv,
    float* __restrict__ out)
{
    const size_t gid = (size_t)blockIdx.x * 256 + threadIdx.x; // group of 4 elems
    const size_t e   = gid * 4;
    const int j0 = (int)(e % Nn);
    const int i  = (int)((e / Nn) % Nn);
    const int b  = (int)(e / ((size_t)Nn * Nn));

    const float di = dinv[(size_t)b * Nn + i];
    const float4 a  = *(const float4*)(adj + e);
    const float4 dj = *(const float4*)(dinv + (size_t)b * Nn + j0);

    float4 r;
    r.x = di * dj.x * (a.x + ((i == j0 + 0) ? 1.0f : 0.0f));
    r.y = di * dj.y * (a.y + ((i == j0 + 1) ? 1.0f : 0.0f));
    r.z = di * dj.z * (a.z + ((i == j0 + 2) ? 1.0f : 0.0f));
    r.w = di * dj.w * (a.w + ((i == j0 + 3) ? 1.0f : 0.0f));
    *(float4*)(out + e) = r;
}

// -------------------------------------------------------------------------
// Kernel 3: edge_embedding[b] = count[b] (512x56) @ emb_table (56x128)
// Full f32 precision via V_WMMA_F32_16X16X4_F32, one wave per 16x16 tile.
// -------------------------------------------------------------------------
__global__ __launch_bounds__(256) void gemm_kernel(
    const float* __restrict__ cnt, const float* __restrict__ emb,
    float* __restrict__ out)
{
    const int wave = threadIdx.x >> 5;
    const int lane = threadIdx.x & 31;
    const int tile = blockIdx.x * 8 + wave;   // 0..255 (32 m-tiles x 8 n-tiles)
    const int b    = blockIdx.y;

    const int mtile = tile >> 3;
    const int ntile = tile & 7;
    const int mbase = mtile * 16;
    const int nbase = ntile * 16;

    const int lhi  = lane >> 4;    // 0: lanes 0-15, 1: lanes 16-31
    const int lrow = lane & 15;

    const float* A = cnt + (((size_t)b * Nn) + mbase) * KDIM; // row-major 16xK tile
    const float* Bm = emb;                                    // 56 x 128 row-major

    v8f acc = {};
    #pragma unroll
    for (int k = 0; k < KDIM; k += 4) {
        v2f a, bb;
        // A 16x4 f32 layout: VGPR0/1 = K {k+2*lhi, k+2*lhi+1}, M = lrow
        a.x = A[lrow * KDIM + k + 2 * lhi + 0];
        a.y = A[lrow * KDIM + k + 2 * lhi + 1];
        // B 4x16 f32 layout (mirror): N = lrow, rows K {k+2*lhi, k+2*lhi+1}
        bb.x = Bm[(k + 2 * lhi + 0) * Uu + nbase + lrow];
        bb.y = Bm[(k + 2 * lhi + 1) * Uu + nbase + lrow];
        acc = __builtin_amdgcn_wmma_f32_16x16x4_f32(
            /*neg_a=*/false, a, /*neg_b=*/false, bb,
            /*c_mod=*/(short)0, acc, /*reuse_a=*/false, /*reuse_b=*/false);
    }

    // D 16x16 f32 layout: VGPR r -> M = mbase + r + 8*lhi, N = nbase + lrow
    float* O = out + ((size_t)b * Nn) * Uu;
    #pragma unroll
    for (int r = 0; r < 8; ++r) {
        const int m = mbase + r + 8 * lhi;
        const int n = nbase + lrow;
        O[(size_t)m * Uu + n] = acc[r];
    }
}

// -------------------------------------------------------------------------
extern "C" void kernel_launch(void* const* d_in, const int* in_sizes, int n_in,
                              void* d_out, int out_size, void* d_ws, size_t ws_size,
                              hipStream_t stream)
{
    const float* loc     = (const float*)d_in[0]; // (B,N)
    const float* inten   = (const float*)d_in[1]; // (B,N)
    const float* pmass   = (const float*)d_in[2]; // (B,)
    const float* mass_aa = (const float*)d_in[3]; // (C,)
    const float* emb     = (const float*)d_in[4]; // (56,U)

    float* out      = (float*)d_out;
    float* adj_norm = out;                                  // B*N*N
    float* edge_emb = out + (size_t)Bb * Nn * Nn;           // B*N*U

    float* adj  = (float*)d_ws;                             // B*N*N
    float* cnt  = adj + (size_t)Bb * Nn * Nn;               // B*N*KDIM
    float* dinv = cnt + (size_t)Bb * Nn * KDIM;             // B*N

    // 1) pairwise scan + histogram + degree
    pair_kernel<<<dim3(Nn, Bb), 256, 0, stream>>>(loc, inten, pmass, mass_aa,
                                                  adj, cnt, dinv);
    // 2) symmetric normalization
    {
        const size_t total4 = ((size_t)Bb * Nn * Nn) / 4;
        norm_kernel<<<dim3((unsigned)(total4 / 256)), 256, 0, stream>>>(adj, dinv, adj_norm);
    }
    // 3) histogram @ emb_table GEMM via f32 WMMA
    gemm_kernel<<<dim3(32, Bb), 256, 0, stream>>>(cnt, emb, edge_emb);
}